// MPNN_12214886990224
// MI455X (gfx1250) — compile-verified
//
#include <hip/hip_runtime.h>

#define E 8192
#define D 256
#define JSPLIT 4

typedef __attribute__((ext_vector_type(2))) float v2f;
typedef __attribute__((ext_vector_type(8))) float v8f;

// D = A(16x4 f32) * B(4x16 f32) + C(16x16 f32), full fp32 on the matrix pipe.
static __device__ __forceinline__ v8f wmma_f32_k4(v2f a, v2f b, v8f c) {
    return __builtin_amdgcn_wmma_f32_16x16x4_f32(
        /*neg_a=*/false, a, /*neg_b=*/false, b,
        /*c_mod=*/(short)0, c, /*reuse_a=*/false, /*reuse_b=*/false);
}

// A-fragment (16x4, MxK) of a row-major matrix at (row0, k0).
// ISA layout: lanes 0-15 -> K={k0,k0+1}, lanes 16-31 -> K={k0+2,k0+3}; M = lane&15.
// This is ALSO the B-fragment of the matrix's transpose (for e @ e^T).
static __device__ __forceinline__ v2f frag_a_rm(const float* __restrict__ p, int ld,
                                                int row0, int k0, int lane) {
    const float* q = p + (size_t)(row0 + (lane & 15)) * ld + k0 + ((lane >> 4) << 1);
    return v2f{q[0], q[1]};
}

// B-fragment (4x16, KxN) of a row-major matrix at (k0, col0).
// lanes 0-15 -> K={k0,k0+1}, lanes 16-31 -> K={k0+2,k0+3}; N = lane&15.
static __device__ __forceinline__ v2f frag_b_rm(const float* __restrict__ p, int ld,
                                                int k0, int col0, int lane) {
    const float* q = p + (size_t)(k0 + ((lane >> 4) << 1)) * ld + col0 + (lane & 15);
    return v2f{q[0], q[ld]};
}

// ---------------------------------------------------------------------------
// Kernel 1: e = relu(X @ W + b) for both sim branches (blockIdx.y selects).
// One wave per block; each wave owns one 16x16 output tile, K=256.
// ---------------------------------------------------------------------------
__global__ __launch_bounds__(32)
void mp_linear_relu_wmma(const float* __restrict__ X,
                         const float* __restrict__ W1, const float* __restrict__ b1,
                         const float* __restrict__ W2, const float* __restrict__ b2,
                         float* __restrict__ e1, float* __restrict__ e2) {
    const int lane = threadIdx.x;
    const int tile = blockIdx.x;
    const int i0 = (tile / (D / 16)) * 16;
    const int n0 = (tile % (D / 16)) * 16;
    const float* W    = blockIdx.y ? W2 : W1;
    const float* bias = blockIdx.y ? b2 : b1;
    float*       out  = blockIdx.y ? e2 : e1;

    v8f c = {};
#pragma unroll 8
    for (int k0 = 0; k0 < D; k0 += 4) {
        v2f a = frag_a_rm(X, D, i0, k0, lane);
        v2f b = frag_b_rm(W, D, k0, n0, lane);
        c = wmma_f32_k4(a, b, c);
    }
    // C/D layout: VGPR r -> row = r + 8*(lane>>4), col = lane&15.
    const int n = lane & 15, half = lane >> 4;
    const float bv = bias[n0 + n];
#pragma unroll
    for (int r = 0; r < 8; ++r) {
        float v = c[r] + bv;
        out[(size_t)(i0 + r + 8 * half) * D + n0 + n] = v > 0.f ? v : 0.f;
    }
}

// ---------------------------------------------------------------------------
// Kernel 2: zero the agg accumulation buffers (needed every call: atomics).
// ---------------------------------------------------------------------------
__global__ __launch_bounds__(256)
void mp_zero_f32(float* __restrict__ p, int n) {
    int i = blockIdx.x * 256 + threadIdx.x;
    if (i < n) p[i] = 0.f;
}

// ---------------------------------------------------------------------------
// Kernel 3: fused agg = ((e @ e^T) / D) @ X without materializing [E,E].
// Grid: (E/16 row-blocks, JSPLIT j-ranges, 2 branches). One wave per block.
// Wave holds the full 16x256 output slab in 128 accumulator VGPRs.
// Per j-tile: 64 WMMA (sim, K=256) + 64 WMMA (agg, 16 N-tiles x K=16) -> no
// recompute waste; sim tile is re-laid C->A through 1KB of LDS.
// ---------------------------------------------------------------------------
__global__ __launch_bounds__(32)
void mp_agg_fused_wmma(const float* __restrict__ X,
                       const float* __restrict__ e1, const float* __restrict__ e2,
                       float* __restrict__ agg1, float* __restrict__ agg2) {
    __shared__ float s_tile[16 * 16];

    const int lane = threadIdx.x;
    const int i0 = blockIdx.x * 16;
    const int jbase = blockIdx.y * (E / JSPLIT);
    const float* e   = blockIdx.z ? e2 : e1;
    float*       agg = blockIdx.z ? agg2 : agg1;

    const int n = lane & 15, half = lane >> 4;
    const float inv_d = 1.0f / (float)D;

    v8f acc[D / 16];
#pragma unroll
    for (int t = 0; t < D / 16; ++t) acc[t] = v8f{};

    for (int j0 = jbase; j0 < jbase + E / JSPLIT; j0 += 16) {
        // ---- stage 1: s = e[i0:,:] @ e[j0:,:]^T  (16x16, K=256) ----
        v8f s = {};
#pragma unroll 8
        for (int k0 = 0; k0 < D; k0 += 4) {
            v2f a = frag_a_rm(e, D, i0, k0, lane);
            v2f b = frag_a_rm(e, D, j0, k0, lane);  // B-frag of e^T == A-frag of e
            s = wmma_f32_k4(a, b, s);
        }
        // scale by 1/D and park in LDS row-major so it can be re-read A-layout
        __syncthreads();
#pragma unroll
        for (int r = 0; r < 8; ++r)
            s_tile[(r + 8 * half) * 16 + n] = s[r] * inv_d;
        __syncthreads();

        // ---- stage 2: acc += s_tile(16x16) @ X[j0:j0+16, :]  (K=16) ----
#pragma unroll
        for (int kc = 0; kc < 4; ++kc) {
            v2f a = frag_a_rm(s_tile, 16, 0, kc * 4, lane);
#pragma unroll
            for (int t = 0; t < D / 16; ++t) {
                v2f b = frag_b_rm(X, D, j0 + kc * 4, t * 16, lane);
                acc[t] = wmma_f32_k4(a, b, acc[t]);
            }
        }
    }

    // accumulate partial slab (one of JSPLIT) into global agg
#pragma unroll
    for (int t = 0; t < D / 16; ++t) {
        float* base = agg + (size_t)(i0 + 8 * half) * D + t * 16 + n;
#pragma unroll
        for (int r = 0; r < 8; ++r)
            atomicAdd(base + (size_t)r * D, acc[t][r]);
    }
}

// ---------------------------------------------------------------------------
// Kernel 4: out = relu(X@Wo[0:256] + agg1@Wo[256:512] + agg2@Wo[512:768] + b).
// One wave per 16x16 output tile, K = 3*256.
// ---------------------------------------------------------------------------
__global__ __launch_bounds__(32)
void mp_out_linear_wmma(const float* __restrict__ X,
                        const float* __restrict__ agg1, const float* __restrict__ agg2,
                        const float* __restrict__ Wo, const float* __restrict__ bo,
                        float* __restrict__ out) {
    const int lane = threadIdx.x;
    const int tile = blockIdx.x;
    const int i0 = (tile / (D / 16)) * 16;
    const int n0 = (tile % (D / 16)) * 16;

    v8f c = {};
    const float* segs[3] = {X, agg1, agg2};
#pragma unroll
    for (int sseg = 0; sseg < 3; ++sseg) {
        const float* A = segs[sseg];
#pragma unroll 8
        for (int k0 = 0; k0 < D; k0 += 4) {
            v2f a = frag_a_rm(A, D, i0, k0, lane);
            v2f b = frag_b_rm(Wo, D, sseg * D + k0, n0, lane);
            c = wmma_f32_k4(a, b, c);
        }
    }
    const int n = lane & 15, half = lane >> 4;
    const float bv = bo[n0 + n];
#pragma unroll
    for (int r = 0; r < 8; ++r) {
        float v = c[r] + bv;
        out[(size_t)(i0 + r + 8 * half) * D + n0 + n] = v > 0.f ? v : 0.f;
    }
}

// ---------------------------------------------------------------------------
extern "C" void kernel_launch(void* const* d_in, const int* in_sizes, int n_in,
                              void* d_out, int out_size, void* d_ws, size_t ws_size,
                              hipStream_t stream) {
    const float* X  = (const float*)d_in[0];   // edge_x [E,D]
    const float* W1 = (const float*)d_in[1];   // W_sim1 [D,D]
    const float* b1 = (const float*)d_in[2];   // b_sim1 [D]
    const float* W2 = (const float*)d_in[3];   // W_sim2 [D,D]
    const float* b2 = (const float*)d_in[4];   // b_sim2 [D]
    const float* Wo = (const float*)d_in[5];   // W_out  [3D,D]
    const float* bo = (const float*)d_in[6];   // b_out  [D]
    float* out = (float*)d_out;

    float* ws   = (float*)d_ws;                // needs 4*E*D floats = 32 MB
    float* e1   = ws;
    float* e2   = ws + (size_t)1 * E * D;
    float* agg1 = ws + (size_t)2 * E * D;
    float* agg2 = ws + (size_t)3 * E * D;

    // 1) e1/e2 = relu(X @ W_sim{1,2} + b)
    mp_linear_relu_wmma<<<dim3((E / 16) * (D / 16), 2), 32, 0, stream>>>(
        X, W1, b1, W2, b2, e1, e2);

    // 2) zero agg accumulators (atomic accumulation target)
    mp_zero_f32<<<(2 * E * D + 255) / 256, 256, 0, stream>>>(agg1, 2 * E * D);

    // 3) fused (e@e^T / D) @ X for both branches
    mp_agg_fused_wmma<<<dim3(E / 16, JSPLIT, 2), 32, 0, stream>>>(
        X, e1, e2, agg1, agg2);

    // 4) out = relu([X, agg1, agg2] @ W_out + b_out)
    mp_out_linear_wmma<<<dim3((E / 16) * (D / 16)), 32, 0, stream>>>(
        X, agg1, agg2, Wo, bo, out);
}